// Net_16561393893886
// MI455X (gfx1250) — compile-verified
//
#include <hip/hip_runtime.h>
#include <hip/hip_bf16.h>
#include <math.h>
#include <stdint.h>

// Problem constants (from reference)
#define NN 50000
#define FF 128
#define GG 3
#define EE 600000

typedef float v2f __attribute__((ext_vector_type(2)));
typedef float v8f __attribute__((ext_vector_type(8)));

// ---------------------------------------------------------------------------
// WMMA GEMM: C[g][rows,128] = A[g][rows,128] @ W[g?][128,128] (+bias) (+=C)
// 256 threads = 8 waves; each wave computes a 16x128 stripe using
// v_wmma_f32_16x16x4_f32.
//
// W is staged into LDS with CDNA5 async global->LDS B32 copies (ASYNCcnt
// path), routed directly into a paired-K layout:
//     lds[((k>>1)*128 + n)*2 + (k&1)] = W[k][n]
// so a B fragment (lanes 0-15: rows k0,k0+1 / lanes 16-31: rows k0+2,k0+3)
// is one aligned 8-byte ds_load_b64 straight into an even VGPR pair --
// no repacking movs in front of the WMMA chain.
// ---------------------------------------------------------------------------
__global__ __launch_bounds__(256)
void wmma_gemm128(const float* __restrict__ A, const float* __restrict__ W,
                  const float* __restrict__ bias, float* __restrict__ C,
                  int nRows, int wGStride, int biasGStride, int accumulate)
{
    __shared__ float ldsW[128 * 128]; // 64 KB, paired-K swizzled
    const int g = blockIdx.y;
    const float*  Wg = W + (size_t)g * wGStride;
    const float*  Ag = A + (size_t)g * nRows * 128;
    float*        Cg = C + (size_t)g * nRows * 128;

    // --- async stage W with per-lane swizzled destinations ---
    {
        uint32_t ldsBase = (uint32_t)(uintptr_t)(void*)ldsW;
#pragma unroll 8
        for (int it = 0; it < 64; ++it) {
            int i = threadIdx.x + it * 256;   // element index 0..16383
            int k = i >> 7, n = i & 127;
            uint32_t l = ldsBase +
                ((((uint32_t)((k >> 1) * 128 + n) << 1) | (uint32_t)(k & 1)) << 2);
            const float* gp = Wg + i;
            asm volatile("global_load_async_to_lds_b32 %0, %1, off"
                         :: "v"(l), "v"(gp) : "memory");
        }
        asm volatile("s_wait_asynccnt 0x0" ::: "memory");
    }
    __syncthreads();

    const int wave = threadIdx.x >> 5;
    const int lane = threadIdx.x & 31;
    const int hi   = lane >> 4;   // 0: lanes 0-15, 1: lanes 16-31
    const int lm   = lane & 15;
    const int row0 = blockIdx.x * 128 + wave * 16;

    v8f acc[8];
#pragma unroll
    for (int t = 0; t < 8; ++t) acc[t] = (v8f)0.0f;

    int rowA = row0 + lm;                 // A-fragment row for this lane
    if (rowA >= nRows) rowA = nRows - 1;  // clamp (stores are guarded)
    const float* aRow = Ag + (size_t)rowA * 128 + (hi << 1);

    // warm the A stripe (global_prefetch_b8)
    __builtin_prefetch(aRow, 0, 0);
    __builtin_prefetch(aRow + 64, 0, 0);

    for (int k0 = 0; k0 < 128; k0 += 4) {
        v2f a = *(const v2f*)(aRow + k0);
        const float* bBase = &ldsW[((((k0 >> 1) + hi) * 128) + lm) << 1];
#pragma unroll
        for (int t = 0; t < 8; ++t) {
            v2f b = *(const v2f*)(bBase + (t << 5)); // +16 cols -> +32 floats
            acc[t] = __builtin_amdgcn_wmma_f32_16x16x4_f32(
                false, a, false, b, (short)0, acc[t], false, false);
        }
    }

    // D layout: VGPR r -> (M=r, N=lane) lanes 0-15 ; (M=r+8, N=lane-16) lanes 16-31
#pragma unroll
    for (int t = 0; t < 8; ++t) {
        int col = t * 16 + lm;
        float bv = bias ? bias[(size_t)g * biasGStride + col] : 0.0f;
#pragma unroll
        for (int r = 0; r < 8; ++r) {
            int row = row0 + r + hi * 8;
            if (row < nRows) {
                size_t idx = (size_t)row * 128 + col;
                float v = acc[t][r] + bv;
                if (accumulate) v += Cg[idx];
                Cg[idx] = v;
            }
        }
    }
}

// ---------------------------------------------------------------------------
// Degree / normalization
// ---------------------------------------------------------------------------
__global__ void deg_init_kernel(float* __restrict__ deg)
{
    int i = blockIdx.x * blockDim.x + threadIdx.x;
    if (i < GG * NN) deg[i] = 1.0f; // self loop
}

__global__ void deg_edge_kernel(const int* __restrict__ ei, float* __restrict__ deg)
{
    long long i = (long long)blockIdx.x * blockDim.x + threadIdx.x;
    if (i >= (long long)GG * EE) return;
    int g = (int)(i / EE);
    int e = (int)(i - (long long)g * EE);
    int d = ei[(size_t)g * 2 * EE + EE + e];
    __hip_atomic_fetch_add(&deg[g * NN + d], 1.0f,
                           __ATOMIC_RELAXED, __HIP_MEMORY_SCOPE_AGENT);
}

__global__ void dinv_kernel(float* __restrict__ deg)
{
    int i = blockIdx.x * blockDim.x + threadIdx.x;
    if (i < GG * NN) deg[i] = rsqrtf(deg[i]); // deg >= 1 always
}

// out[g,n,:] = h[g,n,:] * dinv[g,n]^2 + bias[:]
__global__ void gcn_self_kernel(const float* __restrict__ h,
                                const float* __restrict__ bias,
                                const float* __restrict__ dinv,
                                float* __restrict__ out)
{
    long long i = (long long)blockIdx.x * blockDim.x + threadIdx.x;
    long long total = (long long)GG * NN * 32; // float4 quads
    if (i >= total) return;
    int c = (int)(i & 31);
    long long t = i >> 5;
    int g = (int)(t / NN);
    int n = (int)(t - (long long)g * NN);
    float di = dinv[g * NN + n];
    float w = di * di;
    const float4 hv = *(const float4*)&h[(((size_t)g * NN + n) << 7) + (c << 2)];
    const float4 b4 = *(const float4*)&bias[c << 2];
    float4 o;
    o.x = hv.x * w + b4.x; o.y = hv.y * w + b4.y;
    o.z = hv.z * w + b4.z; o.w = hv.w * w + b4.w;
    *(float4*)&out[(((size_t)g * NN + n) << 7) + (c << 2)] = o;
}

// out[g,d,:] += h[g,s,:] * dinv[s]*dinv[d]  for each edge
__global__ void gcn_scatter_kernel(const float* __restrict__ h,
                                   const int* __restrict__ ei,
                                   const float* __restrict__ dinv,
                                   float* __restrict__ out)
{
    long long total = (long long)GG * EE * 32;
    for (long long i = (long long)blockIdx.x * blockDim.x + threadIdx.x;
         i < total; i += (long long)gridDim.x * blockDim.x) {
        int c = (int)(i & 31);
        long long t = i >> 5;
        int g = (int)(t / EE);
        int e = (int)(t - (long long)g * EE);
        int s = ei[(size_t)g * 2 * EE + e];
        int d = ei[(size_t)g * 2 * EE + EE + e];
        float nrm = dinv[g * NN + s] * dinv[g * NN + d];
        const float4 hv = *(const float4*)&h[(((size_t)g * NN + s) << 7) + (c << 2)];
        float* o = &out[(((size_t)g * NN + d) << 7) + (c << 2)];
        __hip_atomic_fetch_add(o + 0, hv.x * nrm, __ATOMIC_RELAXED, __HIP_MEMORY_SCOPE_AGENT);
        __hip_atomic_fetch_add(o + 1, hv.y * nrm, __ATOMIC_RELAXED, __HIP_MEMORY_SCOPE_AGENT);
        __hip_atomic_fetch_add(o + 2, hv.z * nrm, __ATOMIC_RELAXED, __HIP_MEMORY_SCOPE_AGENT);
        __hip_atomic_fetch_add(o + 3, hv.w * nrm, __ATOMIC_RELAXED, __HIP_MEMORY_SCOPE_AGENT);
    }
}

// ---------------------------------------------------------------------------
// Small reductions / pointwise
// ---------------------------------------------------------------------------
__global__ void zero_kernel(float* __restrict__ p, int count)
{
    for (int i = threadIdx.x; i < count; i += blockDim.x) p[i] = 0.0f;
}

// per-(g,f) column sum of enc over nodes (chunked), atomically into acc[g*128+f]
__global__ void colsum_kernel(const float* __restrict__ enc, float* __restrict__ acc)
{
    int g = blockIdx.y;
    int f = threadIdx.x; // 128 threads
    int n0 = blockIdx.x * 256;
    float s = 0.0f;
    for (int r = 0; r < 256; ++r) {
        int n = n0 + r;
        if (n < NN) s += enc[(((size_t)g * NN + n) << 7) + f];
    }
    __hip_atomic_fetch_add(&acc[g * 128 + f], s,
                           __ATOMIC_RELAXED, __HIP_MEMORY_SCOPE_AGENT);
}

__global__ void gv_final_kernel(const float* __restrict__ acc, float* __restrict__ gv)
{
    int i = blockIdx.x * blockDim.x + threadIdx.x;
    if (i < GG * 128) {
        float m = acc[i] * (1.0f / (float)NN);
        gv[i] = 1.0f / (1.0f + expf(-m));
    }
}

// u[g,d] = sum_e att[g,d,e] * gv[g,e]
__global__ void att_u_kernel(const float* __restrict__ att,
                             const float* __restrict__ gv, float* __restrict__ u)
{
    __shared__ float sg[128];
    int g = blockIdx.x;
    int d = threadIdx.x; // 128
    sg[d] = gv[g * 128 + d];
    __syncthreads();
    const float* row = att + ((size_t)g * 128 + d) * 128;
    float s = 0.0f;
    for (int e = 0; e < 128; ++e) s += row[e] * sg[e];
    u[g * 128 + d] = s;
}

// comp_re[n, i*3+j] = sig(enc[j,n,:].u[i,:]) normalized over j
__global__ void comp_kernel(const float* __restrict__ enc,
                            const float* __restrict__ u,
                            float* __restrict__ comp)
{
    __shared__ float su[3 * 128];
    for (int i = threadIdx.x; i < 3 * 128; i += blockDim.x) su[i] = u[i];
    __syncthreads();
    int n = blockIdx.x * blockDim.x + threadIdx.x;
    if (n >= NN) return;
    float S[3][3];
    for (int j = 0; j < 3; ++j) {
        const float4* ev = (const float4*)&enc[(((size_t)j * NN + n) << 7)];
        float d0 = 0.0f, d1 = 0.0f, d2 = 0.0f;
        for (int c = 0; c < 32; ++c) {
            float4 e4 = ev[c];
            int b = c << 2;
            d0 += e4.x * su[b] + e4.y * su[b + 1] + e4.z * su[b + 2] + e4.w * su[b + 3];
            d1 += e4.x * su[128 + b] + e4.y * su[128 + b + 1] + e4.z * su[128 + b + 2] + e4.w * su[128 + b + 3];
            d2 += e4.x * su[256 + b] + e4.y * su[256 + b + 1] + e4.z * su[256 + b + 2] + e4.w * su[256 + b + 3];
        }
        S[0][j] = 1.0f / (1.0f + expf(-d0));
        S[1][j] = 1.0f / (1.0f + expf(-d1));
        S[2][j] = 1.0f / (1.0f + expf(-d2));
    }
    for (int i = 0; i < 3; ++i) {
        float inv = 1.0f / (S[i][0] + S[i][1] + S[i][2]);
        comp[(size_t)n * 9 + i * 3 + 0] = S[i][0] * inv;
        comp[(size_t)n * 9 + i * 3 + 1] = S[i][1] * inv;
        comp[(size_t)n * 9 + i * 3 + 2] = S[i][2] * inv;
    }
}

// used_feat = elu(mean_g enc)
__global__ void used_feat_kernel(const float* __restrict__ enc, float* __restrict__ out)
{
    long long i = (long long)blockIdx.x * blockDim.x + threadIdx.x;
    if (i >= (long long)NN * 128) return;
    const size_t NF = (size_t)NN * 128;
    float m = (enc[i] + enc[NF + i] + enc[2 * NF + i]) * (1.0f / 3.0f);
    out[i] = m > 0.0f ? m : expm1f(m);
}

// fuse_feat = sigmoid(mean_g each_fin)
__global__ void fuse_feat_kernel(const float* __restrict__ ef, float* __restrict__ out)
{
    long long i = (long long)blockIdx.x * blockDim.x + threadIdx.x;
    if (i >= (long long)NN * 128) return;
    const size_t NF = (size_t)NN * 128;
    float m = (ef[i] + ef[NF + i] + ef[2 * NF + i]) * (1.0f / 3.0f);
    out[i] = 1.0f / (1.0f + expf(-m));
}

// dec_in[g] = (sum_g' enc - enc[g]) / (G-1)
__global__ void dec_in_kernel(const float* __restrict__ enc, float* __restrict__ out)
{
    long long i = (long long)blockIdx.x * blockDim.x + threadIdx.x;
    if (i >= (long long)NN * 128) return;
    const size_t NF = (size_t)NN * 128;
    float a = enc[i], b = enc[NF + i], c = enc[2 * NF + i];
    float s = a + b + c;
    out[i]          = (s - a) * 0.5f;
    out[NF + i]     = (s - b) * 0.5f;
    out[2 * NF + i] = (s - c) * 0.5f;
}

// sum of squared differences over `count` elements, accumulated into acc[gy]
__global__ void sqdiff_kernel(const float* __restrict__ a, const float* __restrict__ b,
                              float* __restrict__ acc, long long count, long long gStride)
{
    __shared__ float red[256];
    const float* ag = a + (size_t)blockIdx.y * gStride;
    const float* bg = b + (size_t)blockIdx.y * gStride;
    float s = 0.0f;
    for (long long i = (long long)blockIdx.x * blockDim.x + threadIdx.x;
         i < count; i += (long long)gridDim.x * blockDim.x) {
        float d = ag[i] - bg[i];
        s += d * d;
    }
    red[threadIdx.x] = s;
    __syncthreads();
    for (int off = 128; off > 0; off >>= 1) {
        if ((int)threadIdx.x < off) red[threadIdx.x] += red[threadIdx.x + off];
        __syncthreads();
    }
    if (threadIdx.x == 0)
        __hip_atomic_fetch_add(&acc[blockIdx.y], red[0],
                               __ATOMIC_RELAXED, __HIP_MEMORY_SCOPE_AGENT);
}

__global__ void finalize_kernel(const float* __restrict__ scal, float* __restrict__ out2)
{
    if (threadIdx.x == 0 && blockIdx.x == 0) {
        float obf1 = 0.0f;
        for (int g = 0; g < GG; ++g)
            obf1 += (sqrtf(scal[g]) + sqrtf(scal[3 + g])) * 0.5f;
        float obf0 = 2.0f * (sqrtf(scal[6]) + sqrtf(scal[7]) + sqrtf(scal[8]));
        out2[0] = obf1;
        out2[1] = obf0;
    }
}

// ---------------------------------------------------------------------------
// Launcher
// ---------------------------------------------------------------------------
extern "C" void kernel_launch(void* const* d_in, const int* in_sizes, int n_in,
                              void* d_out, int out_size, void* d_ws, size_t ws_size,
                              hipStream_t stream)
{
    const float* x       = (const float*)d_in[0];
    const float* fc1_w   = (const float*)d_in[1];
    const float* fc1_b   = (const float*)d_in[2];
    const float* w_conv1 = (const float*)d_in[3];
    const float* b_conv1 = (const float*)d_in[4];
    const float* w_conv2 = (const float*)d_in[5];
    const float* b_conv2 = (const float*)d_in[6];
    const float* w_dc1   = (const float*)d_in[7];
    const float* b_dc1   = (const float*)d_in[8];
    const float* w_dc2   = (const float*)d_in[9];
    const float* b_dc2   = (const float*)d_in[10];
    const float* fc2_w   = (const float*)d_in[11];
    const float* fc2_b   = (const float*)d_in[12];
    const float* att_w   = (const float*)d_in[13];
    const int*   edges   = (const int*)d_in[14];

    const size_t NF  = (size_t)NN * FF;       // per-graph slab elems
    const size_t GNF = (size_t)GG * NF;       // full slab elems

    float* ws    = (float*)d_ws;
    float* pre   = ws + 0 * GNF;
    float* conv1 = ws + 1 * GNF;
    float* enc   = ws + 2 * GNF;
    float* tmp   = ws + 3 * GNF;
    float* slab4 = ws + 4 * GNF; // dec_in, later each_fin
    float* own1  = ws + 5 * GNF;
    float* own2  = ws + 6 * GNF;
    float* oth1  = pre;   // reused after obf1 accumulation
    float* oth2  = conv1; // reused after obf1 accumulation

    float* smallr = ws + 7 * GNF;
    float* dinv   = smallr;                 // G*N
    float* gvacc  = smallr + 150016;        // 384
    float* scal   = gvacc + 384;            // 16
    float* gv     = scal + 16;              // 384
    float* u      = gv + 384;               // 384

    float* out       = (float*)d_out;
    float* out_fuse  = out;
    float* out_used  = out + NF;
    float* out_comp  = out + 2 * NF;
    float* out_scal  = out + 2 * NF + (size_t)NN * 9;

    const dim3 gemmGrid((NN + 127) / 128, GG);
    const dim3 gemmBlk(256);
    const int  elem4   = (int)((GNF / 4 + 255) / 256);   // grid for G*N*32 quads
    const int  elemNF  = (int)((NF + 255) / 256);        // grid for N*128
    const int  scat    = 4096;                           // grid-stride scatter blocks
    const dim3 colGrid((NN + 255) / 256, GG);

    // --- init small accumulators + degree normalization ---
    zero_kernel<<<1, 256, 0, stream>>>(gvacc, 400);
    deg_init_kernel<<<(GG * NN + 255) / 256, 256, 0, stream>>>(dinv);
    deg_edge_kernel<<<(int)(((long long)GG * EE + 255) / 256), 256, 0, stream>>>(edges, dinv);
    dinv_kernel<<<(GG * NN + 255) / 256, 256, 0, stream>>>(dinv);

    // --- pre_feat = x @ fc1_w + fc1_b (per-graph W, per-graph bias) ---
    wmma_gemm128<<<gemmGrid, gemmBlk, 0, stream>>>(x, fc1_w, fc1_b, pre, NN, 128 * 128, 128, 0);

    // --- conv1_f = GCN(pre_feat, w_conv1, b_conv1) ---
    wmma_gemm128<<<gemmGrid, gemmBlk, 0, stream>>>(pre, w_conv1, nullptr, tmp, NN, 0, 0, 0);
    gcn_self_kernel<<<elem4, 256, 0, stream>>>(tmp, b_conv1, dinv, conv1);
    gcn_scatter_kernel<<<scat, 256, 0, stream>>>(tmp, edges, dinv, conv1);

    // --- enc = GCN(conv1_f, w_conv2, b_conv2) ---
    wmma_gemm128<<<gemmGrid, gemmBlk, 0, stream>>>(conv1, w_conv2, nullptr, tmp, NN, 0, 0, 0);
    gcn_self_kernel<<<elem4, 256, 0, stream>>>(tmp, b_conv2, dinv, enc);
    gcn_scatter_kernel<<<scat, 256, 0, stream>>>(tmp, edges, dinv, enc);

    // --- global_vec, u, comp_re, used_feat ---
    colsum_kernel<<<colGrid, 128, 0, stream>>>(enc, gvacc);
    gv_final_kernel<<<2, 256, 0, stream>>>(gvacc, gv);
    att_u_kernel<<<GG, 128, 0, stream>>>(att_w, gv, u);
    comp_kernel<<<(NN + 255) / 256, 256, 0, stream>>>(enc, u, out_comp);
    used_feat_kernel<<<elemNF, 256, 0, stream>>>(enc, out_used);

    // --- dec_in ---
    dec_in_kernel<<<elemNF, 256, 0, stream>>>(enc, slab4);

    // --- own1 = GCN(enc, w_dconv1), own2 = GCN(own1, w_dconv2) ---
    wmma_gemm128<<<gemmGrid, gemmBlk, 0, stream>>>(enc, w_dc1, nullptr, tmp, NN, 0, 0, 0);
    gcn_self_kernel<<<elem4, 256, 0, stream>>>(tmp, b_dc1, dinv, own1);
    gcn_scatter_kernel<<<scat, 256, 0, stream>>>(tmp, edges, dinv, own1);

    wmma_gemm128<<<gemmGrid, gemmBlk, 0, stream>>>(own1, w_dc2, nullptr, tmp, NN, 0, 0, 0);
    gcn_self_kernel<<<elem4, 256, 0, stream>>>(tmp, b_dc2, dinv, own2);
    gcn_scatter_kernel<<<scat, 256, 0, stream>>>(tmp, edges, dinv, own2);

    // --- obf1 accumulation (must precede slab reuse) ---
    sqdiff_kernel<<<dim3(512, GG), 256, 0, stream>>>(pre,   own2, scal + 0, (long long)NF, (long long)NF);
    sqdiff_kernel<<<dim3(512, GG), 256, 0, stream>>>(conv1, own1, scal + 3, (long long)NF, (long long)NF);

    // --- oth1 = GCN(dec_in, w_dconv1), oth2 = GCN(oth1, w_dconv2) ---
    wmma_gemm128<<<gemmGrid, gemmBlk, 0, stream>>>(slab4, w_dc1, nullptr, tmp, NN, 0, 0, 0);
    gcn_self_kernel<<<elem4, 256, 0, stream>>>(tmp, b_dc1, dinv, oth1);
    gcn_scatter_kernel<<<scat, 256, 0, stream>>>(tmp, edges, dinv, oth1);

    wmma_gemm128<<<gemmGrid, gemmBlk, 0, stream>>>(oth1, w_dc2, nullptr, tmp, NN, 0, 0, 0);
    gcn_self_kernel<<<elem4, 256, 0, stream>>>(tmp, b_dc2, dinv, oth2);
    gcn_scatter_kernel<<<scat, 256, 0, stream>>>(tmp, edges, dinv, oth2);

    // --- each_fin = [own2|oth2] @ fc2_w + fc2_b  (two accumulating passes) ---
    wmma_gemm128<<<gemmGrid, gemmBlk, 0, stream>>>(own2, fc2_w,             nullptr, slab4, NN, 0, 0, 0);
    wmma_gemm128<<<gemmGrid, gemmBlk, 0, stream>>>(oth2, fc2_w + 128 * 128, fc2_b,   slab4, NN, 0, 0, 1);

    // --- fuse_feat ---
    fuse_feat_kernel<<<elemNF, 256, 0, stream>>>(slab4, out_fuse);

    // --- obf0 pairwise Frobenius ---
    sqdiff_kernel<<<dim3(512, 1), 256, 0, stream>>>(slab4,      slab4 + NF,     scal + 6, (long long)NF, 0);
    sqdiff_kernel<<<dim3(512, 1), 256, 0, stream>>>(slab4,      slab4 + 2 * NF, scal + 7, (long long)NF, 0);
    sqdiff_kernel<<<dim3(512, 1), 256, 0, stream>>>(slab4 + NF, slab4 + 2 * NF, scal + 8, (long long)NF, 0);

    finalize_kernel<<<1, 32, 0, stream>>>(scal, out_scal);
}